// Base_4621384810644
// MI455X (gfx1250) — compile-verified
//
#include <hip/hip_runtime.h>
#include <math.h>

typedef __attribute__((ext_vector_type(2))) float v2f;
typedef __attribute__((ext_vector_type(8))) float v8f;

// One wave32 per batch element: blocked 64x64 upper-triangular solve with
// 2 RHS. Diagonal 16x16 blocks: shuffle-broadcast back-substitution.
// Off-diagonal block updates: V_WMMA_F32_16X16X4_F32 (full fp32 precision).
//
// RHS lives in a zero-padded, column-major LDS tile RsT[16][65]:
//   RsT[n*65 + row] = column n of the padded 64x16 RHS (cols 0,1 real,
//   cols 2..15 stay zero) -> all WMMA fragment loads/stores are
//   unconditional (no exec-mask branching around ds ops).
__global__ __launch_bounds__(32) void batched_tri_solve_wmma_kernel(
    const float* __restrict__ U_values, const float* __restrict__ V_values,
    const float* __restrict__ mean, const float* __restrict__ mean_post,
    const float* __restrict__ y, const float* __restrict__ noise,
    const int* __restrict__ mini_indices, const int* __restrict__ crow_u,
    const int* __restrict__ crow_v, float* __restrict__ ws)
{
    __shared__ float Vs[64 * 68];    // padded stride 68 (conflict-free col reads)
    __shared__ float RsT[16 * 65];   // transposed, padded RHS (see above)

    const int lane = threadIdx.x;    // 0..31 (wave32)
    const int b    = blockIdx.x;
    const int g    = mini_indices[b];
    const int L    = min(g + 1, 64);
    const int base = 64 - L;         // rows < base are "identity" rows

    // ---- Zero the padded RHS tile
    for (int i = lane; i < 16 * 65; i += 32) RsT[i] = 0.0f;

    // ---- Stage V_sub into LDS (row segments contiguous in V_values -> coalesced)
    for (int p = 0; p < 64; ++p) {
        const bool vrow = (p >= base);
        int rowstart = 0;
        if (vrow) rowstart = crow_v[g - 63 + p];   // anc_j = g-63+p (uniform)
        for (int c = lane; c < 64; c += 32) {
            float val;
            if (vrow && (c >= p)) val = V_values[rowstart + (c - p)]; // anc_i-anc_j = c-p
            else                  val = (c == p) ? 1.0f : 0.0f;       // eye fill
            Vs[p * 68 + c] = val;
        }
    }

    // ---- Stage RHS (e_63 and U_sub) + dot(U_sub, mean - mean_post)
    float dotp = 0.0f;
    const int urow = crow_u[g];
    for (int t = 0; t < 2; ++t) {
        const int p = lane + 32 * t;
        float u = 0.0f, mdv = 0.0f;
        if (p >= base) {
            const int anc = g - 63 + p;
            u   = U_values[urow + (p - base)];
            mdv = mean[anc] - mean_post[anc];
        }
        RsT[0 * 65 + p] = (p == 63) ? 1.0f : 0.0f;
        RsT[1 * 65 + p] = u;
        dotp += u * mdv;
    }
    for (int o = 16; o > 0; o >>= 1) dotp += __shfl_down(dotp, o, 32); // lane0
    __syncthreads();

    const int h  = lane >> 4;   // lane half
    const int lm = lane & 15;

    // ---- Blocked back-substitution: block rows kb = 3 (bottom) .. 0 (top)
    #pragma unroll
    for (int kb = 3; kb >= 0; --kb) {
        // Diagonal 16x16 solve: lanes 0..15 own rows, 16..31 mirror harmlessly.
        const int m = kb * 16 + lm;
        float r0 = RsT[0 * 65 + m];
        float r1 = RsT[1 * 65 + m];
        const float dinv = 1.0f / Vs[m * 68 + m];   // 1.0 on identity rows
        #pragma unroll
        for (int j = 15; j >= 0; --j) {
            if (lm == j) { r0 *= dinv; r1 *= dinv; }
            const float x0 = __shfl(r0, j, 32);
            const float x1 = __shfl(r1, j, 32);
            if (lm < j) {
                const float vij = Vs[m * 68 + kb * 16 + j];
                r0 -= vij * x0;
                r1 -= vij * x1;
            }
        }
        if (h == 0) { RsT[0 * 65 + m] = r0; RsT[1 * 65 + m] = r1; } // X_kb -> LDS
        __syncthreads();

        // Off-diagonal updates R_k -= A_{k,kb} * X_kb  (4x chained WMMA f32 16x16x4).
        // Fragment rule (ISA 7.12.2): VGPR v, lane half h, lane-in-half lm:
        //   A[m][k]: a[v] = A[lm][4s + v + 2h]
        //   B[k][n]: b[v] = B[4s + v + 2h][lm]   (contiguous in RsT -> b64)
        //   C[m][n]: c[v] = C[v + 8h][lm]
        #pragma unroll
        for (int k = 0; k < kb; ++k) {
            const int cbase = lm * 65 + 16 * k + 8 * h;
            v8f c;
            #pragma unroll
            for (int v = 0; v < 8; ++v) c[v] = RsT[cbase + v];

            #pragma unroll
            for (int s = 0; s < 4; ++s) {
                const int acol = 16 * kb + 4 * s + 2 * h;
                v2f a;
                a.x = Vs[(16 * k + lm) * 68 + acol + 0];
                a.y = Vs[(16 * k + lm) * 68 + acol + 1];
                const int xrow = lm * 65 + 16 * kb + 4 * s + 2 * h;
                v2f bb;           // -X packed (f32 WMMA NEG only supports C-negate)
                bb.x = -RsT[xrow + 0];
                bb.y = -RsT[xrow + 1];
                // D = A*(-X) + C  ==  C - A*X
                c = __builtin_amdgcn_wmma_f32_16x16x4_f32(
                        false, a, false, bb, (short)0, c, false, false);
            }
            #pragma unroll
            for (int v = 0; v < 8; ++v) RsT[cbase + v] = c[v];  // cols>=2 stay 0
        }
        __syncthreads();
    }

    // ---- Norms of the two solutions
    float s0 = 0.0f, s1 = 0.0f;
    for (int t = 0; t < 2; ++t) {
        const int p = lane + 32 * t;
        const float a = RsT[0 * 65 + p];
        const float q = RsT[1 * 65 + p];
        s0 += a * a;
        s1 += q * q;
    }
    for (int o = 16; o > 0; o >>= 1) {
        s0 += __shfl_down(s0, o, 32);
        s1 += __shfl_down(s1, o, 32);
    }

    if (lane == 0) {
        const float ulast  = U_values[crow_u[g + 1] - 1];
        const float vfirst = V_values[crow_v[g]];
        const float resid  = y[g] - mean_post[g];
        const float ns     = noise[0];
        // logDet_b + innerMean_b + innerCov_b - (resid^2 + marginalVar_b)/(2*noise)
        const float contrib = logf(ulast) - logf(vfirst)
                            - 0.5f * dotp * dotp
                            - 0.5f * s1
                            - (resid * resid + s0) / (2.0f * ns);
        ws[b] = contrib;
    }
}

// Deterministic fixed-order reduction of per-block partials + constant term.
__global__ __launch_bounds__(256) void final_reduce_kernel(
    const float* __restrict__ ws, const float* __restrict__ noise,
    float* __restrict__ out, int B)
{
    __shared__ float sm[256];
    float p = 0.0f;
    for (int i = threadIdx.x; i < B; i += 256) p += ws[i];
    sm[threadIdx.x] = p;
    __syncthreads();
    for (int o = 128; o > 0; o >>= 1) {
        if ((int)threadIdx.x < o) sm[threadIdx.x] += sm[threadIdx.x + o];
        __syncthreads();
    }
    if (threadIdx.x == 0) {
        const float ns = noise[0];
        out[0] = sm[0] - 0.5f * (float)B * logf(6.283185307179586f * ns);
    }
}

extern "C" void kernel_launch(void* const* d_in, const int* in_sizes, int n_in,
                              void* d_out, int out_size, void* d_ws, size_t ws_size,
                              hipStream_t stream) {
    const float* U         = (const float*)d_in[0];
    const float* V         = (const float*)d_in[1];
    const float* mean      = (const float*)d_in[2];
    const float* mean_post = (const float*)d_in[3];
    const float* y         = (const float*)d_in[4];
    const float* noise     = (const float*)d_in[5];
    const int*   mini      = (const int*)d_in[6];
    const int*   crow_u    = (const int*)d_in[7];
    const int*   crow_v    = (const int*)d_in[8];
    const int B = in_sizes[6];

    float* ws = (float*)d_ws;   // B floats of per-block partials

    batched_tri_solve_wmma_kernel<<<B, 32, 0, stream>>>(
        U, V, mean, mean_post, y, noise, mini, crow_u, crow_v, ws);
    final_reduce_kernel<<<1, 256, 0, stream>>>(ws, noise, (float*)d_out, B);
}